// ocr_roi_pooling_78048145703389
// MI455X (gfx1250) — compile-verified
//
#include <hip/hip_runtime.h>

// OCR ROI max-pooling for MI455X (gfx1250, wave32).
// feats: (B=4, C=128, H=64, W=512) fp32
// rois : (N, 5) fp32 [batch, x1, y1, x2, y2] in image coords
// out  : (N, C, PH=8, PW=32) fp32
//
// Design notes (compile-only reasoning from MI455X specs):
//  * 64 MB feature map is L2-resident (192 MB L2) -> latency-bound gather.
//  * wave32 mapping: each wave = one ph row, lanes = pw -> coalesced stores,
//    spatially local gathers per wave.
//  * 4-way channel unroll -> 4 independent loads in flight per window elem.
//  * global_prefetch_b8 (gfx1250) warms next channel group.

#define PH_  8
#define PW_  32
#define C_   128
#define H_   64
#define W_   512
#define HSS_ 0.25f
#define WSS_ 0.25f
#define MAXBIN_ (H_ / PH_ + 2)   // 10, replicates reference truncation
#define NEG_ (-1e+37f)
#define CPB_ 16                  // channels per block (grid.y = C_/CPB_ = 8)

__global__ __launch_bounds__(256)
void ocr_roi_pool_kernel(const float* __restrict__ feats,
                         const float* __restrict__ rois,
                         float* __restrict__ out) {
    const int n  = blockIdx.x;
    const int c0 = blockIdx.y * CPB_;
    const int t  = threadIdx.x;
    const int pw = t & 31;   // lane id within wave32
    const int ph = t >> 5;   // wave id within block

    // ---- ROI decode (uniform per block -> scalar loads/ALU) ----
    const float* r = rois + (size_t)n * 5;
    const int rb  = (int)r[0];
    const int rsw = (int)floorf(r[1] * WSS_ + 0.5f);
    const int rsh = (int)floorf(r[2] * HSS_ + 0.5f);
    const int rew = (int)floorf(r[3] * WSS_ + 0.5f);
    const int reh = (int)floorf(r[4] * HSS_ + 0.5f);
    const int roi_w = max(rew - rsw + 1, 1);
    const int roi_h = max(reh - rsh + 1, 1);
    const int rpw   = (PH_ * roi_w + roi_h - 1) / roi_h;
    const float bsh = (float)roi_h / (float)PH_;
    const float bsw = (float)roi_w / (float)rpw;

    // ---- per-thread bin boundaries (match reference clipping exactly) ----
    int hs = (int)floorf((float)ph * bsh) + rsh;
    int he = (int)ceilf((float)(ph + 1) * bsh) + rsh;
    hs = min(max(hs, 0), H_);
    he = min(max(he, 0), H_);
    int ws = (int)floorf((float)pw * bsw) + rsw;
    int we = (int)ceilf((float)(pw + 1) * bsw) + rsw;
    ws = min(max(ws, 0), W_);
    we = min(max(we, 0), W_);

    const bool pad   = (ws >= rew);
    const bool empty = (he <= hs) || (we <= ws);
    const bool zero  = empty || pad;
    // reference only scans MAXBIN_ offsets in each dimension
    const int he_c = min(he, hs + MAXBIN_);
    const int we_c = min(we, ws + MAXBIN_);

    const float* fb = feats + (size_t)rb * C_ * H_ * W_;
    float* ob = out + (((size_t)n * C_ + c0) * PH_ + ph) * PW_ + pw;

    for (int cc = 0; cc < CPB_; cc += 4) {
        const int c = c0 + cc;
        const float* p0 = fb + (size_t)c * (H_ * W_);
        const float* p1 = p0 + (H_ * W_);
        const float* p2 = p1 + (H_ * W_);
        const float* p3 = p2 + (H_ * W_);

        // gfx1250: warm next channel group's first needed row while we reduce
        if (cc + 4 < CPB_) {
            __builtin_prefetch(fb + (size_t)(c + 4) * (H_ * W_) + hs * W_ + ws, 0, 3);
        }

        float a0 = NEG_, a1 = NEG_, a2 = NEG_, a3 = NEG_;
        for (int h = hs; h < he_c; ++h) {
            const int row = h * W_;
            #pragma unroll 2
            for (int w = ws; w < we_c; ++w) {
                const int idx = row + w;
                a0 = fmaxf(a0, p0[idx]);   // 4 independent L2 loads in flight
                a1 = fmaxf(a1, p1[idx]);
                a2 = fmaxf(a2, p2[idx]);
                a3 = fmaxf(a3, p3[idx]);
            }
        }
        if (zero) { a0 = 0.0f; a1 = 0.0f; a2 = 0.0f; a3 = 0.0f; }

        // lanes = pw -> each wave stores a contiguous 128B line
        ob[(size_t)(cc + 0) * (PH_ * PW_)] = a0;
        ob[(size_t)(cc + 1) * (PH_ * PW_)] = a1;
        ob[(size_t)(cc + 2) * (PH_ * PW_)] = a2;
        ob[(size_t)(cc + 3) * (PH_ * PW_)] = a3;
    }
}

extern "C" void kernel_launch(void* const* d_in, const int* in_sizes, int n_in,
                              void* d_out, int out_size, void* d_ws, size_t ws_size,
                              hipStream_t stream) {
    const float* feats = (const float*)d_in[0];
    const float* rois  = (const float*)d_in[1];
    float* out = (float*)d_out;

    const int nroi = in_sizes[1] / 5;   // 256 in the reference setup
    dim3 grid(nroi, C_ / CPB_);         // (256, 8) blocks
    dim3 block(256);                    // 8 waves of 32 (PH_ x PW_ plane)
    ocr_roi_pool_kernel<<<grid, block, 0, stream>>>(feats, rois, out);
}